// ObservationModel2D_76055280878227
// MI455X (gfx1250) — compile-verified
//
#include <hip/hip_runtime.h>

// 2D uniform-grid binning of N particles into (nx*ny) cells.
// Strategy: per-block LDS-privatized int histogram (64KB for 128x128; CDNA5
// WGP has 320KB LDS), streamed non-temporal 128-bit loads unrolled x2 for
// memory-level parallelism, unconditional speculative global_prefetch, then
// each block dumps its private histogram to workspace with plain coalesced
// stores; a reduction kernel sums partials with ZERO global atomics.
// Fallback path (tiny ws or nbins > LDS capacity): direct global f32 atomics.

#define NBINS_LDS 16384
#define HIST_TPB  1024   // 32 wave32 waves per block

typedef __attribute__((ext_vector_type(4))) float f4;

__global__ void zero_f32_kernel(float* __restrict__ out, int n) {
  int i = blockIdx.x * blockDim.x + threadIdx.x;
  if (i < n) out[i] = 0.0f;
}

// ---------------- Path 1: LDS-privatized partial histograms ----------------

__global__ __launch_bounds__(HIST_TPB) void hist_partial_kernel(
    const float* __restrict__ particles, int npart,
    const float* __restrict__ lo, const float* __restrict__ hi,
    const int* __restrict__ nxp, const int* __restrict__ nyp,
    int* __restrict__ partials)
{
  __shared__ int hist[NBINS_LDS];

  const int nx = nxp[0];
  const int ny = nyp[0];
  const int nbins = nx * ny;            // host guarantees nbins <= NBINS_LDS
  const float lo0 = lo[0], lo1 = lo[1];
  const float hi0 = hi[0], hi1 = hi[1];
  const float cx = (hi0 - lo0) / (float)nx;   // cell widths (reference formula)
  const float cy = (hi1 - lo1) / (float)ny;

  const int tid = threadIdx.x;
  for (int b = tid; b < nbins; b += HIST_TPB) hist[b] = 0;
  __syncthreads();

  const int gid    = blockIdx.x * HIST_TPB + tid;
  const int gsize  = gridDim.x * HIST_TPB;
  const int npairs = npart >> 1;        // two particles per 128-bit load
  const f4* __restrict__ pp = (const f4*)particles;

  // bin both particles of one 128-bit load into the LDS histogram
  auto bin2 = [&](f4 v) {
#pragma unroll
    for (int k = 0; k < 2; ++k) {
      const float px = k ? v.z : v.x;
      const float py = k ? v.w : v.y;
      if (px >= lo0 && px < hi0 && py >= lo1 && py < hi1) {
        int ix = (int)floorf((px - lo0) / cx);
        int iy = (int)floorf((py - lo1) / cy);
        ix = min(max(ix, 0), nx - 1);
        iy = min(max(iy, 0), ny - 1);
        atomicAdd(&hist[ix * ny + iy], 1);   // ds_add_u32, 64-bank LDS
      }
    }
  };

  int i = gid;
  // unrolled x2: two independent b128 loads in flight per iteration
  for (; i + gsize < npairs; i += 2 * gsize) {
    f4 a = __builtin_nontemporal_load(&pp[i]);
    f4 b = __builtin_nontemporal_load(&pp[i + gsize]);
    // speculative prefetch one unroll ahead (OOB-safe: silently dropped)
    __builtin_prefetch(&pp[i + 2 * gsize], 0, 0);
    __builtin_prefetch(&pp[i + 3 * gsize], 0, 0);
    bin2(a);
    bin2(b);
  }
  for (; i < npairs; i += gsize) {
    f4 a = __builtin_nontemporal_load(&pp[i]);
    bin2(a);
  }

  // odd-count tail: one thread handles the last particle
  if ((npart & 1) && gid == 0) {
    const float px = particles[(size_t)(npart - 1) * 2 + 0];
    const float py = particles[(size_t)(npart - 1) * 2 + 1];
    if (px >= lo0 && px < hi0 && py >= lo1 && py < hi1) {
      int ix = (int)floorf((px - lo0) / cx);
      int iy = (int)floorf((py - lo1) / cy);
      ix = min(max(ix, 0), nx - 1);
      iy = min(max(iy, 0), ny - 1);
      atomicAdd(&hist[ix * ny + iy], 1);
    }
  }

  __syncthreads();
  // Plain coalesced stores of the private histogram (no global atomics).
  int* __restrict__ mypart = partials + (size_t)blockIdx.x * nbins;
  for (int b = tid; b < nbins; b += HIST_TPB)
    mypart[b] = hist[b];
}

__global__ void reduce_partials_kernel(const int* __restrict__ partials,
                                       int nbins, int nparts,
                                       float* __restrict__ out)
{
  int b = blockIdx.x * blockDim.x + threadIdx.x;
  if (b >= nbins) return;
  int s = 0;
  for (int p = 0; p < nparts; ++p)
    s += partials[(size_t)p * nbins + b];   // coalesced across threads
  out[b] = (float)s;                        // exact: total 8M < 2^24
}

// ---------------- Path 2 (fallback): direct global float atomics ----------

__global__ __launch_bounds__(256) void hist_direct_kernel(
    const float* __restrict__ particles, int npart,
    const float* __restrict__ lo, const float* __restrict__ hi,
    const int* __restrict__ nxp, const int* __restrict__ nyp,
    float* __restrict__ out)
{
  const int nx = nxp[0];
  const int ny = nyp[0];
  const float lo0 = lo[0], lo1 = lo[1];
  const float hi0 = hi[0], hi1 = hi[1];
  const float cx = (hi0 - lo0) / (float)nx;
  const float cy = (hi1 - lo1) / (float)ny;

  const int gid    = blockIdx.x * blockDim.x + threadIdx.x;
  const int gsize  = gridDim.x * blockDim.x;
  const int npairs = npart >> 1;
  const f4* __restrict__ pp = (const f4*)particles;

  for (int i = gid; i < npairs; i += gsize) {
    f4 v = __builtin_nontemporal_load(&pp[i]);
#pragma unroll
    for (int k = 0; k < 2; ++k) {
      const float px = k ? v.z : v.x;
      const float py = k ? v.w : v.y;
      if (px >= lo0 && px < hi0 && py >= lo1 && py < hi1) {
        int ix = (int)floorf((px - lo0) / cx);
        int iy = (int)floorf((py - lo1) / cy);
        ix = min(max(ix, 0), nx - 1);
        iy = min(max(iy, 0), ny - 1);
        atomicAdd(&out[ix * ny + iy], 1.0f);   // global_atomic_add_f32
      }
    }
  }
  if ((npart & 1) && gid == 0) {
    const float px = particles[(size_t)(npart - 1) * 2 + 0];
    const float py = particles[(size_t)(npart - 1) * 2 + 1];
    if (px >= lo0 && px < hi0 && py >= lo1 && py < hi1) {
      int ix = (int)floorf((px - lo0) / cx);
      int iy = (int)floorf((py - lo1) / cy);
      ix = min(max(ix, 0), nx - 1);
      iy = min(max(iy, 0), ny - 1);
      atomicAdd(&out[ix * ny + iy], 1.0f);
    }
  }
}

extern "C" void kernel_launch(void* const* d_in, const int* in_sizes, int n_in,
                              void* d_out, int out_size, void* d_ws, size_t ws_size,
                              hipStream_t stream) {
  const float* particles = (const float*)d_in[0];
  const float* lo        = (const float*)d_in[1];
  const float* hi        = (const float*)d_in[2];
  const int*   nxp       = (const int*)d_in[3];
  const int*   nyp       = (const int*)d_in[4];
  float* out = (float*)d_out;

  const int npart = in_sizes[0] / 2;   // particles is (N,2) flat
  const int nbins = out_size;          // nx*ny

  // How many private partial histograms fit in workspace?
  const size_t per_partial = (size_t)nbins * sizeof(int);
  int B = 256;                         // target: 256 blocks x 32 waves
  if (per_partial > 0 && (size_t)B * per_partial > ws_size)
    B = (int)(ws_size / per_partial);

  if (nbins <= NBINS_LDS && B >= 8) {
    // Main path: LDS private histograms -> ws partials -> atomic-free reduce.
    int* partials = (int*)d_ws;
    hist_partial_kernel<<<B, HIST_TPB, 0, stream>>>(
        particles, npart, lo, hi, nxp, nyp, partials);
    int rblocks = (nbins + 255) / 256;
    reduce_partials_kernel<<<rblocks, 256, 0, stream>>>(partials, nbins, B, out);
  } else {
    // Fallback: zero output, then direct global float atomics.
    zero_f32_kernel<<<(out_size + 255) / 256, 256, 0, stream>>>(out, out_size);
    hist_direct_kernel<<<512, 256, 0, stream>>>(
        particles, npart, lo, hi, nxp, nyp, out);
  }
}